// QuantumMLP_3255585210674
// MI455X (gfx1250) — compile-verified
//
#include <hip/hip_runtime.h>
#include <math.h>

#define TWO_PI_F 6.28318530717958647692f

typedef float v2f __attribute__((ext_vector_type(2)));
typedef float v8f __attribute__((ext_vector_type(8)));

// ---------------------------------------------------------------------------
// Prep kernel: half-angle sin/cos of the (batch-uniform) variational params.
// trig layout (floats): [0..11]=cos(rz/2) [12..23]=sin(rz/2)
//                       [24..35]=cos(ry/2) [36..47]=sin(ry/2)
// ---------------------------------------------------------------------------
__global__ void qmlp_prep_kernel(const float* __restrict__ rz,
                                 const float* __restrict__ ry,
                                 float* __restrict__ trig) {
  int t = threadIdx.x;
  if (t < 12) {
    float hz = 0.5f * rz[t];
    float hy = 0.5f * ry[t];
    trig[t]      = cosf(hz);
    trig[12 + t] = sinf(hz);
    trig[24 + t] = cosf(hy);
    trig[36 + t] = sinf(hy);
  }
}

__device__ __forceinline__ float lane_bperm(float v, int srcLane) {
  return __int_as_float(__builtin_amdgcn_ds_bpermute(srcLane << 2, __float_as_int(v)));
}

__device__ __forceinline__ float fast_sigmoid(float x) {
  // v_exp_f32 + v_rcp_f32 (~1 ulp each) instead of libm expf + IEEE divide.
  return __builtin_amdgcn_rcpf(1.0f + __expf(-x));
}

// Wire w (0..2) acts on bit (4 >> w) of the flat index q0*4 + q1*2 + q2.
__device__ __forceinline__ void gate_rz(float sr[8], float si[8], int w,
                                        float cm, float sm) {
  const int st = 4 >> w;
#pragma unroll
  for (int i = 0; i < 8; ++i) {
    if ((i & st) == 0) {
      int j = i | st;
      // a0 *= (cm - i sm); a1 *= (cm + i sm)
      float r0 = sr[i]*cm + si[i]*sm;
      float i0 = si[i]*cm - sr[i]*sm;
      float r1 = sr[j]*cm - si[j]*sm;
      float i1 = si[j]*cm + sr[j]*sm;
      sr[i]=r0; si[i]=i0; sr[j]=r1; si[j]=i1;
    }
  }
}

__device__ __forceinline__ void gate_ry(float sr[8], float si[8], int w,
                                        float c, float s) {
  const int st = 4 >> w;
#pragma unroll
  for (int i = 0; i < 8; ++i) {
    if ((i & st) == 0) {
      int j = i | st;
      float r0 = c*sr[i] - s*sr[j];
      float i0 = c*si[i] - s*si[j];
      float r1 = s*sr[i] + c*sr[j];
      float i1 = s*si[i] + c*si[j];
      sr[i]=r0; si[i]=i0; sr[j]=r1; si[j]=i1;
    }
  }
}

__device__ __forceinline__ void gate_cnot(float sr[8], float si[8], int c, int t) {
  const int cb = 4 >> c, tb = 4 >> t;
#pragma unroll
  for (int i = 0; i < 8; ++i) {
    if ((i & cb) != 0 && (i & tb) == 0) {
      int j = i | tb;
      float tr=sr[i]; sr[i]=sr[j]; sr[j]=tr;
      float ti=si[i]; si[i]=si[j]; si[j]=ti;
    }
  }
}

// ---------------------------------------------------------------------------
// Main kernel: one sample per lane; 8 complex amps in 16 VGPR floats.
// Head (z @ W^T + b) done per-wave with V_WMMA_F32_16X16X4_F32 (fp32, K=3->4).
// ---------------------------------------------------------------------------
__global__ __launch_bounds__(256) void qmlp_main_kernel(
    const float* __restrict__ x,
    const float* __restrict__ trig,
    const float* __restrict__ W,      // [3,3] row-major
    const float* __restrict__ bias,   // [3]
    float* __restrict__ out,          // [B,3]
    int B)
{
  const int gid  = blockIdx.x * 256 + threadIdx.x;
  const int sidx = gid < B ? gid : (B - 1);   // clamp (keeps EXEC full)

  float x0 = x[3*sidx+0], x1 = x[3*sidx+1], x2 = x[3*sidx+2];

  // Spherical encoding
  float theta = acosf(x2);
  float phi   = fmodf(atan2f(x1, x0) + TWO_PI_F, TWO_PI_F);
  float cth, sth, cph, sph;
  sincosf(0.5f*theta, &sth, &cth);
  sincosf(0.5f*phi,   &sph, &cph);

  // Single-qubit amps after RY(theta), RZ(phi) on |0>; identical on all wires,
  // so initial state is the 3-fold Kronecker product.
  float u0r = cth*cph, u0i = -cth*sph;
  float u1r = sth*cph, u1i =  sth*sph;
  float p00r = u0r*u0r - u0i*u0i, p00i = 2.0f*u0r*u0i;
  float p01r = u0r*u1r - u0i*u1i, p01i = u0r*u1i + u0i*u1r;
  float p11r = u1r*u1r - u1i*u1i, p11i = 2.0f*u1r*u1i;

  float sr[8], si[8];
  sr[0]=p00r*u0r-p00i*u0i; si[0]=p00r*u0i+p00i*u0r;
  sr[1]=p00r*u1r-p00i*u1i; si[1]=p00r*u1i+p00i*u1r;
  sr[2]=p01r*u0r-p01i*u0i; si[2]=p01r*u0i+p01i*u0r;
  sr[3]=p01r*u1r-p01i*u1i; si[3]=p01r*u1i+p01i*u1r;
  sr[4]=sr[2];             si[4]=si[2];
  sr[5]=sr[3];             si[5]=si[3];
  sr[6]=p11r*u0r-p11i*u0i; si[6]=p11r*u0i+p11i*u0r;
  sr[7]=p11r*u1r-p11i*u1i; si[7]=p11r*u1i+p11i*u1r;

  // Variational layers (params are wave-uniform scalar loads)
#pragma unroll
  for (int l = 0; l < 4; ++l) {
#pragma unroll
    for (int w = 0; w < 3; ++w) {
      int k = l*3 + w;
      gate_rz(sr, si, w, trig[k],      trig[12+k]);
      gate_ry(sr, si, w, trig[24+k],   trig[36+k]);
    }
    gate_cnot(sr, si, 0, 1);
    gate_cnot(sr, si, 1, 2);
    gate_cnot(sr, si, 2, 0);
  }

  // <Z_w> expectation values
  float p[8];
#pragma unroll
  for (int i = 0; i < 8; ++i) p[i] = sr[i]*sr[i] + si[i]*si[i];
  float z0 = (p[0]+p[1]+p[2]+p[3]) - (p[4]+p[5]+p[6]+p[7]);
  float z1 = (p[0]+p[1]+p[4]+p[5]) - (p[2]+p[3]+p[6]+p[7]);
  float z2 = (p[0]+p[2]+p[4]+p[6]) - (p[1]+p[3]+p[5]+p[7]);

  // ---- head: sigmoid(z @ W^T + b) via V_WMMA_F32_16X16X4_F32 -------------
  const int  lane = threadIdx.x & 31;
  const int  n    = lane & 15;          // output column owned in B/C/D
  const bool hi   = lane >= 16;
  const int  waveBase = blockIdx.x * 256 + (threadIdx.x & ~31);

  float Wn0 = 0.f, Wn1 = 0.f, Wn2 = 0.f, bn = 0.f;
  if (n < 3) { Wn0 = W[n*3+0]; Wn1 = W[n*3+1]; Wn2 = W[n*3+2]; bn = bias[n]; }

  // B (4x16): VGPR0 = {K=0: W[n][0] | K=2: W[n][2]}, VGPR1 = {K=1: W[n][1] | K=3: 0}
  v2f Bm; Bm.x = hi ? Wn2 : Wn0; Bm.y = hi ? 0.0f : Wn1;
  // C: bias broadcast to every row
  v8f Cm;
#pragma unroll
  for (int r = 0; r < 8; ++r) Cm[r] = bn;

  // A group 0 (samples waveBase+0..15): hi lanes carry K=2 (z2 of lane-16)
  float z2lo = lane_bperm(z2, lane & 15);
  v2f A0; A0.x = hi ? z2lo : z0; A0.y = hi ? 0.0f : z1;
  // A group 1 (samples waveBase+16..31): lo lanes carry z0/z1 of lane+16
  float z0hi = lane_bperm(z0, (lane + 16) & 31);
  float z1hi = lane_bperm(z1, (lane + 16) & 31);
  v2f A1; A1.x = hi ? z2 : z0hi; A1.y = hi ? 0.0f : z1hi;

  v8f D0 = __builtin_amdgcn_wmma_f32_16x16x4_f32(false, A0, false, Bm,
                                                 (short)0, Cm, false, false);
  v8f D1 = __builtin_amdgcn_wmma_f32_16x16x4_f32(false, A1, false, Bm,
                                                 (short)0, Cm, false, false);

  // D layout: VGPR r, lane -> row (r + 8*hi), col n; only cols 0..2 are real.
  if (n < 3) {
    const int mofs = hi ? 8 : 0;
    const int base = (waveBase + mofs) * 3 + n;   // out index of (row 0, col n)
    if (waveBase + 32 <= B) {
      // Fast path (always taken for B % 32 == 0): unguarded, immediate-offset
      // stores; group 1 is +16 samples = +48 floats.
#pragma unroll
      for (int r = 0; r < 8; ++r) {
        out[base + 3*r]      = fast_sigmoid(D0[r]);
        out[base + 3*r + 48] = fast_sigmoid(D1[r]);
      }
    } else {
      // Ragged tail (never hit for B = 1<<20, kept for safety).
#pragma unroll
      for (int r = 0; r < 8; ++r) {
        int s0 = waveBase + mofs + r;
        int s1 = s0 + 16;
        if (s0 < B) out[s0*3 + n] = fast_sigmoid(D0[r]);
        if (s1 < B) out[s1*3 + n] = fast_sigmoid(D1[r]);
      }
    }
  }
}

// ---------------------------------------------------------------------------
extern "C" void kernel_launch(void* const* d_in, const int* in_sizes, int n_in,
                              void* d_out, int out_size, void* d_ws, size_t ws_size,
                              hipStream_t stream) {
  const float* x    = (const float*)d_in[0];   // [B,3]
  const float* rz_p = (const float*)d_in[1];   // [4,3]
  const float* ry_p = (const float*)d_in[2];   // [4,3]
  const float* W    = (const float*)d_in[3];   // [3,3]
  const float* b    = (const float*)d_in[4];   // [3]
  float* out  = (float*)d_out;                 // [B,3]
  float* trig = (float*)d_ws;                  // 48 floats scratch

  const int B = in_sizes[0] / 3;
  qmlp_prep_kernel<<<1, 32, 0, stream>>>(rz_p, ry_p, trig);
  const int grid = (B + 255) / 256;
  qmlp_main_kernel<<<grid, 256, 0, stream>>>(x, trig, W, b, out, B);
}